// Word2BoxCBOW_80453327388837
// MI455X (gfx1250) — compile-verified
//
#include <hip/hip_runtime.h>
#include <cstdint>

#define NUM_BOXES 100001
#define TWO_GAMMA 1.1544313298030658f   // 2 * EulerGamma * BETA
#define LV_EPS    1e-23f
#define LOG2E     1.4426950408889634f
#define LN2       0.6931471805599453f

typedef float v2f __attribute__((ext_vector_type(2)));
typedef float v8f __attribute__((ext_vector_type(8)));

// log1p(exp(-|d|)) via raw v_exp_f32 / v_log_f32 (base-2 formulation).
// exp2 arg <= 0 -> result in (0,1]; log2 arg in (1,2] -> always normal range,
// so the OCML denormal-guard + extended-precision ln2 scaffolding is unnecessary.
__device__ __forceinline__ float softlog(float d) {
    float t = __builtin_amdgcn_exp2f(-fabsf(d) * LOG2E);
    return LN2 * __builtin_amdgcn_logf(1.0f + t);
}
// gmax(a,b) = logaddexp(a,b) = max + log1p(e^-|a-b|)  (>= max always)
__device__ __forceinline__ float gmaxf2(float a, float b) {
    return fmaxf(a, b) + softlog(a - b);
}
// gmin(a,b) = -logaddexp(-a,-b) = min - log1p(e^-|a-b|)
__device__ __forceinline__ float gminf2(float a, float b) {
    return fminf(a, b) - softlog(a - b);
}
// log(softplus(z) + eps), softplus(z) = max(z,0) + log1p(e^-|z|).
// sp + eps >= 1e-23 ~ 2^-77: normal float, raw v_log_f32 is safe.
__device__ __forceinline__ float lvterm(float z) {
    float sp = fmaxf(z, 0.0f) + softlog(z);
    return LN2 * __builtin_amdgcn_logf(sp + LV_EPS);
}

__device__ __forceinline__ unsigned fix_idx(int v) {
    return ((unsigned)(v + NUM_BOXES)) % (unsigned)NUM_BOXES;
}

__global__ __launch_bounds__(256) void
w2b_cbow_kernel(const int* __restrict__ x,
                const float* __restrict__ cen_lo, const float* __restrict__ cen_hi,
                const float* __restrict__ ctx_lo, const float* __restrict__ ctx_hi,
                float* __restrict__ out, int Brows)
{
    __shared__ int xsh[8][336];                 // 8 waves * 16 rows * 21 ints

    const int lane = threadIdx.x & 31;
    const int w    = threadIdx.x >> 5;
    const int tile = blockIdx.x * 8 + w;        // 16 rows per wave-tile
    const int ntiles = Brows >> 4;
    if (tile >= ntiles) return;                 // uniform per wave -> EXEC stays all-1s

    const int rowBase = tile * 16;
    const int rl   = lane & 15;                 // row within tile  (M of WMMA)
    const int half = lane >> 4;                 // selects K in {0,1} vs {2,3}
    const int d0   = 2 * half;                  // dim offset within a K=4 chunk

    // ---- async-stage this tile's 336 indices (1344 contiguous bytes) into LDS ----
    const int* xg = x + (size_t)rowBase * 21;
    unsigned ldsBase = (unsigned)(uintptr_t)(&xsh[w][0]);
    for (int c = lane; c < 84; c += 32) {       // 84 x b128 chunks
        unsigned la = ldsBase + (unsigned)c * 16u;
        const int* gp = xg + c * 4;
        asm volatile("global_load_async_to_lds_b128 %0, %1, off"
                     :: "v"(la), "v"(gp) : "memory");
    }
    __builtin_prefetch(xg + 336, 0, 1);         // next tile's indices -> global_prefetch_b8
    asm volatile("s_wait_asynccnt 0x0" ::: "memory");

    const int* myx = &xsh[w][rl * 21];

    // ---- fold the 10 context boxes: acc_lo = gmax-fold, acc_hi = gmin-fold ----
    float2 accL[16], accH[16];
    {
        unsigned i0 = fix_idx(myx[11]);
        const float2* pl = (const float2*)(ctx_lo + (size_t)i0 * 64 + d0);
        const float2* ph = (const float2*)(ctx_hi + (size_t)i0 * 64 + d0);
#pragma unroll
        for (int j = 0; j < 16; ++j) { accL[j] = pl[2 * j]; accH[j] = ph[2 * j]; }
    }
    for (int i = 1; i < 10; ++i) {
        unsigned ii = fix_idx(myx[11 + i]);
        const float2* ql = (const float2*)(ctx_lo + (size_t)ii * 64 + d0);
        const float2* qh = (const float2*)(ctx_hi + (size_t)ii * 64 + d0);
#pragma unroll
        for (int j = 0; j < 16; ++j) {
            float2 bl = ql[2 * j], bh = qh[2 * j];
            accL[j].x = gmaxf2(accL[j].x, bl.x);
            accL[j].y = gmaxf2(accL[j].y, bl.y);
            accH[j].x = gminf2(accH[j].x, bh.x);
            accH[j].y = gminf2(accH[j].y, bh.y);
        }
    }

    // ---- 11 candidates (t=0: positive, t=1..10: negatives), all from cen tables ----
    const v2f ones = {1.0f, 1.0f};              // B matrix = all ones -> WMMA row-sum
    for (int t = 0; t < 11; ++t) {
        unsigned ci = fix_idx(myx[t]);
        const float2* ql = (const float2*)(cen_lo + (size_t)ci * 64 + d0);
        const float2* qh = (const float2*)(cen_hi + (size_t)ci * 64 + d0);

        v8f acc = {};                           // 16x16 f32 accumulator (row sums)
#pragma unroll
        for (int j = 0; j < 16; ++j) {          // 16 chained K=4 WMMAs cover D=64
            float2 bl = ql[2 * j], bh = qh[2 * j];
            float lo0 = gmaxf2(bl.x, accL[j].x);
            float lo1 = gmaxf2(bl.y, accL[j].y);
            float hi0 = gminf2(bh.x, accH[j].x);
            float hi1 = gminf2(bh.y, accH[j].y);
            v2f a;                              // A fragment: lane rl = row, K = d0,d0+1
            a.x = lvterm(hi0 - lo0 - TWO_GAMMA);
            a.y = lvterm(hi1 - lo1 - TWO_GAMMA);
            acc = __builtin_amdgcn_wmma_f32_16x16x4_f32(
                      false, a, false, ones, (short)0, acc, false, false);
        }

        // D layout: row m -> VGPR m&7 on lane-half m>>3 (all N columns identical).
        // Lane 0 writes rows 0..7, lane 16 writes rows 8..15.
        if (rl == 0) {
            int mbase = half * 8;
#pragma unroll
            for (int v = 0; v < 8; ++v) {
                int r = rowBase + mbase + v;
                float s = acc[v];
                if (t == 0) out[r] = s;                       // positive_score (B,1)
                else        out[Brows + r * 10 + (t - 1)] = s; // negative_score (B,10)
            }
        }
    }
}

extern "C" void kernel_launch(void* const* d_in, const int* in_sizes, int n_in,
                              void* d_out, int out_size, void* d_ws, size_t ws_size,
                              hipStream_t stream) {
    const int*   x      = (const int*)d_in[0];
    const float* cen_lo = (const float*)d_in[1];
    const float* cen_hi = (const float*)d_in[2];
    const float* ctx_lo = (const float*)d_in[3];
    const float* ctx_hi = (const float*)d_in[4];
    float* out = (float*)d_out;

    int Brows  = in_sizes[0] / 21;          // 32768 rows of (1 + 10 + 10) indices
    int ntiles = Brows / 16;                // 2048 wave-tiles
    int blocks = (ntiles + 7) / 8;          // 8 waves (256 threads) per block

    w2b_cbow_kernel<<<blocks, 256, 0, stream>>>(x, cen_lo, cen_hi, ctx_lo, ctx_hi,
                                                out, Brows);
}